// SinkhornDivergenceLoss_17386027614699
// MI455X (gfx1250) — compile-verified
//
#include <hip/hip_runtime.h>
#include <hip/hip_bf16.h>
#include <math.h>

typedef __attribute__((ext_vector_type(2))) float v2f;
typedef __attribute__((ext_vector_type(4))) float v4f;
typedef __attribute__((ext_vector_type(8))) float v8f;

#define BATCH 8
#define NPTS  1024
#define DIM   32
#define TILES (NPTS / 16)            // 64 column tiles
#define WAVES 4
#define TPW   (TILES / WAVES)        // 16 tiles per wave

// ---- cross-lane reductions (wave32) ------------------------------------
// masks <= 8 never cross the 16-lane halves, so these reduce within each
// 16-lane group (matching the 16x16 C/D layout: N = lane % 16).
__device__ inline float rmax16(float v) {
  v = fmaxf(v, __shfl_xor(v, 1));
  v = fmaxf(v, __shfl_xor(v, 2));
  v = fmaxf(v, __shfl_xor(v, 4));
  v = fmaxf(v, __shfl_xor(v, 8));
  return v;
}
__device__ inline float rsum16(float v) {
  v += __shfl_xor(v, 1);
  v += __shfl_xor(v, 2);
  v += __shfl_xor(v, 4);
  v += __shfl_xor(v, 8);
  return v;
}
__device__ inline float rsum32(float v) {
  v += __shfl_xor(v, 1);
  v += __shfl_xor(v, 2);
  v += __shfl_xor(v, 4);
  v += __shfl_xor(v, 8);
  v += __shfl_xor(v, 16);
  return v;
}

// ---- squared row norms for x and y -------------------------------------
__global__ void norms_kernel(const float* __restrict__ x, const float* __restrict__ y,
                             float* __restrict__ xn2, float* __restrict__ yn2) {
  int idx = blockIdx.x * blockDim.x + threadIdx.x;
  if (idx >= 2 * BATCH * NPTS) return;
  const float* row;
  float* out;
  if (idx < BATCH * NPTS) { row = x + (size_t)idx * DIM; out = xn2 + idx; }
  else { int k = idx - BATCH * NPTS; row = y + (size_t)k * DIM; out = yn2 + k; }
  const v4f* r4 = (const v4f*)row;
  float s = 0.f;
#pragma unroll
  for (int q = 0; q < DIM / 4; ++q) {
    v4f v = r4[q];
    s += v.x * v.x + v.y * v.y + v.z * v.z + v.w * v.w;
  }
  *out = s;
}

// ---- fused WMMA tile + streaming logsumexp pass ------------------------
// outvec[b,i] = logconst - logsumexp_col( rscale*rvec[col] - D[i,col] )
// D[i,col] = normA[i] + normB[col] - 2*<Arow_i, Brow_col>
// A fragments are pre-scaled by -2 so the WMMA result is already -2*dot.
__global__ __launch_bounds__(128) void lse_pass(
    const float* __restrict__ Arows, const float* __restrict__ Brows,
    const float* __restrict__ normA, const float* __restrict__ normB,
    const float* __restrict__ rvec, float rscale,
    float* __restrict__ outvec, float logconst) {
  const int b    = blockIdx.y;
  const int i0   = blockIdx.x * 16;
  const int tid  = threadIdx.x;
  const int wid  = tid >> 5;
  const int lane = tid & 31;
  const int half = lane >> 4;   // 0: K=0,1 ; 1: K=2,3 per 4-wide chunk
  const int lml  = lane & 15;

  // A fragment (16 rows x 32 d), preloaded for all 8 K-chunks, scaled by -2.
  const float* arow = Arows + ((size_t)b * NPTS + i0 + lml) * DIM;
  v2f afrag[8];
#pragma unroll
  for (int kk = 0; kk < 8; ++kk) {
    v2f a = *(const v2f*)(arow + 4 * kk + 2 * half);
    afrag[kk] = -2.f * a;
  }

  float na[8];
#pragma unroll
  for (int v = 0; v < 8; ++v)
    na[v] = normA[(size_t)b * NPTS + i0 + v + 8 * half];

  // Per-lane streaming logsumexp state (no shuffles inside the loop).
  float m[8], s[8];
#pragma unroll
  for (int v = 0; v < 8; ++v) { m[v] = -INFINITY; s[v] = 0.f; }

  const size_t colbase = (size_t)b * NPTS + wid * TPW * 16 + lml;
  const float* brow0 = Brows + colbase * DIM;

  // Software pipeline: prefetch tile t+1's B fragments during tile t's WMMAs.
  v2f bcur[8];
#pragma unroll
  for (int kk = 0; kk < 8; ++kk)
    bcur[kk] = *(const v2f*)(brow0 + 4 * kk + 2 * half);
  float nb = normB[colbase];
  float rv = rscale * rvec[colbase];

  for (int t = 0; t < TPW; ++t) {
    const int tn = (t + 1 < TPW) ? (t + 1) : t;   // branchless last-iter clamp
    const float* brown = brow0 + (size_t)tn * 16 * DIM;
    v2f bnext[8];
#pragma unroll
    for (int kk = 0; kk < 8; ++kk)
      bnext[kk] = *(const v2f*)(brown + 4 * kk + 2 * half);
    const float nbn = normB[colbase + (size_t)tn * 16];
    const float rvn = rscale * rvec[colbase + (size_t)tn * 16];

    // Two independent accumulator chains for ILP.
    v8f acc0 = {}, acc1 = {};
#pragma unroll
    for (int kk = 0; kk < 8; kk += 2) {
      acc0 = __builtin_amdgcn_wmma_f32_16x16x4_f32(
          false, afrag[kk], false, bcur[kk], (short)0, acc0, false, false);
      acc1 = __builtin_amdgcn_wmma_f32_16x16x4_f32(
          false, afrag[kk + 1], false, bcur[kk + 1], (short)0, acc1, false, false);
    }

    const float t1 = rv - nb;
#pragma unroll
    for (int v = 0; v < 8; ++v) {
      float kv = t1 - na[v] + (acc0[v] + acc1[v]);  // rv - D
      float nm = fmaxf(m[v], kv);
      s[v] = s[v] * __expf(m[v] - nm) + __expf(kv - nm);
      m[v] = nm;
    }

#pragma unroll
    for (int kk = 0; kk < 8; ++kk) bcur[kk] = bnext[kk];
    nb = nbn;
    rv = rvn;
  }

  // Merge per-lane states across the 16-lane group (once).
  __shared__ float lm[WAVES][16];
  __shared__ float ls[WAVES][16];
#pragma unroll
  for (int v = 0; v < 8; ++v) {
    float nm = rmax16(m[v]);
    float S  = rsum16(s[v] * __expf(m[v] - nm));
    if (lml == 0) { lm[wid][v + 8 * half] = nm; ls[wid][v + 8 * half] = S; }
  }
  __syncthreads();
  if (tid < 16) {
    float M = -INFINITY, S = 0.f;
#pragma unroll
    for (int w = 0; w < WAVES; ++w) {
      float mm = lm[w][tid], ss = ls[w][tid];
      float nm = fmaxf(M, mm);
      S = S * __expf(M - nm) + ss * __expf(mm - nm);
      M = nm;
    }
    outvec[(size_t)b * NPTS + i0 + tid] = logconst - (M + __logf(S));
  }
}

// ---- fused transport-plan loss accumulation ----------------------------
// partial[b*TILES + band] = sum over band's 16 rows x all cols of
//   exp(r_i + c_j - D_ij) * D_ij
__global__ __launch_bounds__(128) void loss_pass(
    const float* __restrict__ x, const float* __restrict__ y,
    const float* __restrict__ xn2, const float* __restrict__ yn2,
    const float* __restrict__ rvec, const float* __restrict__ cvec,
    float* __restrict__ partial) {
  const int b    = blockIdx.y;
  const int band = blockIdx.x;
  const int i0   = band * 16;
  const int tid  = threadIdx.x;
  const int wid  = tid >> 5;
  const int lane = tid & 31;
  const int half = lane >> 4;
  const int lml  = lane & 15;

  const float* arow = x + ((size_t)b * NPTS + i0 + lml) * DIM;
  v2f afrag[8];
#pragma unroll
  for (int kk = 0; kk < 8; ++kk) {
    v2f a = *(const v2f*)(arow + 4 * kk + 2 * half);
    afrag[kk] = -2.f * a;
  }

  float na[8], rr[8];
#pragma unroll
  for (int v = 0; v < 8; ++v) {
    na[v] = xn2[(size_t)b * NPTS + i0 + v + 8 * half];
    rr[v] = rvec[(size_t)b * NPTS + i0 + v + 8 * half];
  }

  const size_t colbase = (size_t)b * NPTS + wid * TPW * 16 + lml;
  const float* brow0 = y + colbase * DIM;

  v2f bcur[8];
#pragma unroll
  for (int kk = 0; kk < 8; ++kk)
    bcur[kk] = *(const v2f*)(brow0 + 4 * kk + 2 * half);
  float nb = yn2[colbase];
  float cv = cvec[colbase];

  float accsum = 0.f;
  for (int t = 0; t < TPW; ++t) {
    const int tn = (t + 1 < TPW) ? (t + 1) : t;
    const float* brown = brow0 + (size_t)tn * 16 * DIM;
    v2f bnext[8];
#pragma unroll
    for (int kk = 0; kk < 8; ++kk)
      bnext[kk] = *(const v2f*)(brown + 4 * kk + 2 * half);
    const float nbn = yn2[colbase + (size_t)tn * 16];
    const float cvn = cvec[colbase + (size_t)tn * 16];

    v8f acc0 = {}, acc1 = {};
#pragma unroll
    for (int kk = 0; kk < 8; kk += 2) {
      acc0 = __builtin_amdgcn_wmma_f32_16x16x4_f32(
          false, afrag[kk], false, bcur[kk], (short)0, acc0, false, false);
      acc1 = __builtin_amdgcn_wmma_f32_16x16x4_f32(
          false, afrag[kk + 1], false, bcur[kk + 1], (short)0, acc1, false, false);
    }

#pragma unroll
    for (int v = 0; v < 8; ++v) {
      float Dv = na[v] + nb + (acc0[v] + acc1[v]);
      accsum = __fmaf_rn(__expf(rr[v] + cv - Dv), Dv, accsum);
    }

#pragma unroll
    for (int kk = 0; kk < 8; ++kk) bcur[kk] = bnext[kk];
    nb = nbn;
    cv = cvn;
  }

  float wsum = rsum32(accsum);
  __shared__ float lsum[WAVES];
  if (lane == 0) lsum[wid] = wsum;
  __syncthreads();
  if (tid == 0)
    partial[(size_t)b * TILES + band] = lsum[0] + lsum[1] + lsum[2] + lsum[3];
}

// ---- deterministic fixed-order final reduction -------------------------
__global__ void final_reduce(const float* __restrict__ partial, float* __restrict__ out) {
  __shared__ float sm[TILES];
  const int b = blockIdx.x;
  const int t = threadIdx.x;
  sm[t] = partial[(size_t)b * TILES + t];
  __syncthreads();
  for (int off = TILES / 2; off > 0; off >>= 1) {
    if (t < off) sm[t] += sm[t + off];
    __syncthreads();
  }
  if (t == 0) out[b] = sm[0];
}

extern "C" void kernel_launch(void* const* d_in, const int* in_sizes, int n_in,
                              void* d_out, int out_size, void* d_ws, size_t ws_size,
                              hipStream_t stream) {
  const float* x = (const float*)d_in[0];
  const float* y = (const float*)d_in[1];
  float* out = (float*)d_out;

  float* ws      = (float*)d_ws;
  float* xn2     = ws;                      // B*N
  float* yn2     = xn2 + BATCH * NPTS;      // B*N
  float* rvec    = yn2 + BATCH * NPTS;      // B*N
  float* cvec    = rvec + BATCH * NPTS;     // B*N
  float* partial = cvec + BATCH * NPTS;     // B*TILES

  const float logu = logf(1.0f / (float)NPTS + 1e-10f);

  norms_kernel<<<(2 * BATCH * NPTS + 255) / 256, 256, 0, stream>>>(x, y, xn2, yn2);

  dim3 grid(TILES, BATCH);
  // Pass A: r_i = logu - logsumexp_j(-D_ij)      (rows = x, cols = y)
  lse_pass<<<grid, 128, 0, stream>>>(x, y, xn2, yn2, xn2, 0.0f, rvec, logu);
  // Pass B: c_j = logu - logsumexp_i(r_i - D_ij) (rows = y, cols = x, +r bias)
  lse_pass<<<grid, 128, 0, stream>>>(y, x, yn2, xn2, rvec, 1.0f, cvec, logu);
  // Pass C: per-band partial of sum(T * D)
  loss_pass<<<grid, 128, 0, stream>>>(x, y, xn2, yn2, rvec, cvec, partial);
  // Deterministic final sum -> d_out[b]
  final_reduce<<<BATCH, TILES, 0, stream>>>(partial, out);
}